// LRBlockAttentionBlock_51926154609237
// MI455X (gfx1250) — compile-verified
//
#include <hip/hip_runtime.h>
#include <math.h>

typedef __attribute__((ext_vector_type(2))) float v2f;
typedef __attribute__((ext_vector_type(8))) float v8f;

// ---------------------------------------------------------------------------
// Problem constants (B=4, H=W=128, block 8x8 -> 1024 blocks of 64 tokens)
// ---------------------------------------------------------------------------
#define NTOK   64
#define FSTR   23      // fb row stride (22 ch + pad)
#define GSTR   25      // normalized/scaled feature stride (24 cols + pad)
#define SSTR   65      // scores stride (64 + pad)
#define DH     176     // irreps-H dimension (8*9 + 8*13)

// LDS layout (floats), regions aliased across phases:
#define OFF_FB   0                       // 64*23  raw features         (all phases)
#define OFF_SH   (OFF_FB + NTOK*FSTR)    // 64*6   sh per token         (all phases)
#define OFF_PB   (OFF_SH + NTOK*6)       // 64     positional bias
#define OFF_SC   (OFF_PB + NTOK)         // 64*65  scores -> attn
#define OFF_E    (OFF_SC + NTOK*SSTR)    // 14744: g(ph1-2) -> vals(ph4-5) -> cgOut+wto(ph6)
#define OFF_CTX  (OFF_E + 14744)         // 11264: wtv+cgVal staging(ph0-4) -> ctx(ph5-6)
#define OFF_DL   (OFF_CTX + NTOK*DH)     // 64*23  delta
#define SMEM_FLOATS (OFF_DL + NTOK*FSTR) // 33560 floats = 134240 B  (<320KB/WGP)

// CG tensors in workspace (float offsets). tp_val set first, tp_out second.
#define CG_VAL_TOTAL 2670
#define CG_OUT_TOTAL 10648

// ---------------------------------------------------------------------------
// Init kernel: compute normalized real-basis Wigner-3j tensors on device
// ---------------------------------------------------------------------------
struct CgSpec { int l1, l2, l3, off; };
__constant__ CgSpec g_cgspec[14] = {
    // tp_val CGs
    {4,0,4,     0}, {4,2,4,    81}, {4,2,6,   486},
    {6,0,6,  1071}, {6,2,4,  1240}, {6,2,6,  1825},
    // tp_out CGs (base 2670)
    {4,4,4,  2670}, {4,4,6,  3399}, {4,6,4,  4452}, {4,6,6,  5505},
    {6,4,4,  7026}, {6,4,6,  8079}, {6,6,4,  9600}, {6,6,6, 11121},
};

__device__ double d_fact(int n) {
    double r = 1.0;
    for (int i = 2; i <= n; ++i) r *= (double)i;
    return r;
}

__device__ double su2_cg(int j1, int j2, int j3, int m1, int m2, int m3) {
    if (m1 + m2 != m3) return 0.0;
    int vmin = max(max(-j1 + j2 + m3, -j1 + m1), 0);
    int vmax = min(min(j2 + j3 + m1, j3 - j1 + j2), j3 + m3);
    if (vmax < vmin) return 0.0;
    double c = sqrt((2.0*j3+1.0) * d_fact(j3+j1-j2) * d_fact(j3-j1+j2) *
                    d_fact(j1+j2-j3) / d_fact(j1+j2+j3+1))
             * sqrt(d_fact(j3+m3)*d_fact(j3-m3) /
                    (d_fact(j1+m1)*d_fact(j1-m1)*d_fact(j2+m2)*d_fact(j2-m2)));
    double s = 0.0;
    for (int v = vmin; v <= vmax; ++v) {
        double t = d_fact(j2+j3+m1-v) * d_fact(j1-m1+v) /
                   (d_fact(v)*d_fact(j3-j1+j2-v)*d_fact(j3+m3-v)*d_fact(v+j1-j2-m3));
        s += (((v + j2 + m2) & 1) ? -1.0 : 1.0) * t;
    }
    return c * s;
}

// q(l)[row][col] entry (without the (-i)^l phase); q has <=2 nonzeros per column
__device__ void q_entry(int l, int row, int col, double& re, double& im) {
    const double is2 = 0.70710678118654752440;
    re = 0.0; im = 0.0;
    int m = row - l, p = col - l;
    if (p == 0) { if (m == 0) re = 1.0; return; }
    int q = p < 0 ? -p : p;
    int am = m < 0 ? -m : m;
    if (am != q) return;
    if (m > 0) {
        double sgn = (m & 1) ? -1.0 : 1.0;
        if (p > 0) re = sgn * is2; else im = sgn * is2;
    } else {
        if (p > 0) re = is2; else im = -is2;
    }
}

__device__ int cand_rows(int l, int col, int rows[2]) {
    int p = col - l;
    if (p == 0) { rows[0] = l; return 1; }
    int q = p < 0 ? -p : p;
    rows[0] = l + q; rows[1] = l - q;
    return 2;
}

// out[a,b,c] = Re( q1[:,a] x q2[:,b] x conj(q3[:,c]) . su2_cg ), phase applied later
__device__ double cg_element(int l1, int l2, int l3, int a, int b, int c) {
    int r1[2], r2[2], r3[2];
    int n1 = cand_rows(l1, a, r1);
    int n2 = cand_rows(l2, b, r2);
    int n3 = cand_rows(l3, c, r3);
    double acc = 0.0;
    for (int ii = 0; ii < n1; ++ii)
        for (int kk = 0; kk < n2; ++kk)
            for (int nn = 0; nn < n3; ++nn) {
                int m1 = r1[ii]-l1, m2 = r2[kk]-l2, m3 = r3[nn]-l3;
                if (m1 + m2 != m3) continue;
                double cgv = su2_cg(l1, l2, l3, m1, m2, m3);
                if (cgv == 0.0) continue;
                double ar, ai, br, bi, cr, ci;
                q_entry(l1, r1[ii], a, ar, ai);
                q_entry(l2, r2[kk], b, br, bi);
                q_entry(l3, r3[nn], c, cr, ci);
                ci = -ci; // conj(q3)
                double rr = ar*br - ai*bi;
                double ri = ar*bi + ai*br;
                acc += (rr*cr - ri*ci) * cgv;
            }
    return acc;
}

__global__ __launch_bounds__(256)
void cg_init_kernel(float* __restrict__ ws) {
    __shared__ double red[256];
    CgSpec sp = g_cgspec[blockIdx.x];
    int d1 = 2*sp.l1+1, d2 = 2*sp.l2+1, d3 = 2*sp.l3+1;
    int n = d1*d2*d3;
    // (-i)^l for even l is real: +1 if l%4==0 else -1. conj((-i)^l3) == same (real).
    double ph = ((sp.l1 % 4 == 0) ? 1.0 : -1.0) *
                ((sp.l2 % 4 == 0) ? 1.0 : -1.0) *
                ((sp.l3 % 4 == 0) ? 1.0 : -1.0);
    double v[9];
    double ssq = 0.0;
    int cnt = 0;
    for (int e = threadIdx.x; e < n; e += 256) {
        int a = e / (d2*d3), r = e % (d2*d3);
        double t = cg_element(sp.l1, sp.l2, sp.l3, a, r / d3, r % d3);
        v[cnt++] = t;
        ssq += t*t;
    }
    red[threadIdx.x] = ssq;
    __syncthreads();
    for (int s = 128; s > 0; s >>= 1) {
        if (threadIdx.x < s) red[threadIdx.x] += red[threadIdx.x + s];
        __syncthreads();
    }
    double scale = ph / sqrt(red[0]);
    cnt = 0;
    for (int e = threadIdx.x; e < n; e += 256)
        ws[sp.off + e] = (float)(v[cnt++] * scale);
}

// ---------------------------------------------------------------------------
// Main kernel: one workgroup per 8x8 spatial block (1024 blocks, 256 threads)
// ---------------------------------------------------------------------------
__global__ __launch_bounds__(256)
void lrblock_attn_kernel(const float* __restrict__ feat,
                         const float* __restrict__ sh_block,
                         const float* __restrict__ log_s4,
                         const float* __restrict__ log_s6,
                         const float* __restrict__ pb_w,
                         const float* __restrict__ pb_b,
                         const float* __restrict__ w_in4,
                         const float* __restrict__ w_in6,
                         const float* __restrict__ w_tp_val,
                         const float* __restrict__ w_tp_out,
                         const float* __restrict__ w_out4,
                         const float* __restrict__ w_out6,
                         const float* __restrict__ cg,
                         float* __restrict__ out) {
    extern __shared__ float sm[];
    const int tid  = threadIdx.x;
    const int lane = tid & 31;
    const int wv   = tid >> 5;
    const int lr   = lane & 15;   // M/N index within WMMA tile
    const int lh   = lane >> 4;   // lane half -> K pair select
    const int bb   = blockIdx.x;
    const int bimg = bb >> 8;
    const int by   = (bb >> 4) & 15;
    const int bx   = bb & 15;

    // ---- Phase 0: stage features / sh / tp_val weights / tp_val CGs --------
    for (int i = tid; i < NTOK * 22; i += 256) {
        int t = i / 22, ch = i % 22;
        int n = (by*8 + (t >> 3)) * 128 + (bx*8 + (t & 7));
        sm[OFF_FB + t*FSTR + ch] = feat[(size_t)(bimg*16384 + n)*22 + ch];
    }
    for (int i = tid; i < NTOK*6; i += 256) sm[OFF_SH + i] = sh_block[i];
    for (int i = tid; i < 384;    i += 256) sm[OFF_CTX + i] = w_tp_val[i];        // wtv
    for (int i = tid; i < CG_VAL_TOTAL; i += 256) sm[OFF_CTX + 384 + i] = cg[i];  // cgVal
    __syncthreads();

    // ---- Phase 1: normalize + fold sqrt(scale); positional bias ------------
    if (tid < NTOK) {
        const int t = tid;
        float s4 = __expf(0.5f * log_s4[0]);
        float s6 = __expf(0.5f * log_s6[0]);
        float n4 = 0.f, n6 = 0.f;
        for (int i = 0; i < 9;  ++i) { float x = sm[OFF_FB + t*FSTR + i];     n4 += x*x; }
        for (int i = 0; i < 13; ++i) { float x = sm[OFF_FB + t*FSTR + 9 + i]; n6 += x*x; }
        float r4 = s4 / fmaxf(sqrtf(n4), 1e-12f);
        float r6 = s6 / fmaxf(sqrtf(n6), 1e-12f);
        for (int i = 0; i < 9;  ++i) sm[OFF_E + t*GSTR + i]     = sm[OFF_FB + t*FSTR + i]     * r4;
        for (int i = 0; i < 13; ++i) sm[OFF_E + t*GSTR + 9 + i] = sm[OFF_FB + t*FSTR + 9 + i] * r6;
        sm[OFF_E + t*GSTR + 22] = 0.f;
        sm[OFF_E + t*GSTR + 23] = 0.f;
        float p = pb_b[0];
        for (int j = 0; j < 6; ++j) p += sm[OFF_SH + t*6 + j] * pb_w[j];
        sm[OFF_PB + t] = p;
    }
    __syncthreads();

    // ---- Phase 2: scores = G G^T (64x64, K=24) via V_WMMA_F32_16X16X4_F32 --
    for (int tt = wv*2; tt < wv*2 + 2; ++tt) {
        const int mt = tt >> 2, nt = tt & 3;
        v8f acc = {};
        for (int kk = 0; kk < 6; ++kk) {
            int k0 = kk*4 + lh*2;
            v2f A, Bv;
            A.x  = sm[OFF_E + (mt*16 + lr)*GSTR + k0];
            A.y  = sm[OFF_E + (mt*16 + lr)*GSTR + k0 + 1];
            Bv.x = sm[OFF_E + (nt*16 + lr)*GSTR + k0];     // B = G_nt^T
            Bv.y = sm[OFF_E + (nt*16 + lr)*GSTR + k0 + 1];
            acc = __builtin_amdgcn_wmma_f32_16x16x4_f32(false, A, false, Bv,
                                                        (short)0, acc, false, false);
        }
        for (int e = 0; e < 8; ++e) {
            int row = mt*16 + e + 8*lh, col = nt*16 + lr;
            sm[OFF_SC + row*SSTR + col] = acc[e] + sm[OFF_PB + row] + sm[OFF_PB + col];
        }
    }
    __syncthreads();

    // ---- Phase 3: row softmax ----------------------------------------------
    if (tid < NTOK) {
        float mx = -3.0e38f;
        for (int m = 0; m < 64; ++m) mx = fmaxf(mx, sm[OFF_SC + tid*SSTR + m]);
        float sum = 0.f;
        for (int m = 0; m < 64; ++m) {
            float e = __expf(sm[OFF_SC + tid*SSTR + m] - mx);
            sm[OFF_SC + tid*SSTR + m] = e;
            sum += e;
        }
        float inv = 1.0f / sum;
        for (int m = 0; m < 64; ++m) sm[OFF_SC + tid*SSTR + m] *= inv;
    }
    __syncthreads();

    // ---- Phase 4: vals = TP(h, sh); h's channel dep. factors out -----------
    // Register arrays only ever use compile-time indices; the long 'i' loop
    // stays rolled and streams f[i] from LDS (no scratch spills).
    {
        const float PWV4 = 0.61237243570f;   // sqrt(9/24)
        const float PWV6 = 0.73598007219f;   // sqrt(13/24)
        for (int task = tid; task < 512; task += 256) {
            const int t = task >> 3, u = task & 7;
            float shv[6];
            #pragma unroll
            for (int j = 0; j < 6; ++j) shv[j] = sm[OFF_SH + t*6 + j];
            float o4[9] = {0}, o6[13] = {0};
            const int L1[6] = {4,4,4,6,6,6};
            const int L2[6] = {0,2,2,0,2,2};
            const int LO[6] = {4,4,6,6,4,6};
            const int CO[6] = {0,81,486,1071,1240,1825};
            #pragma unroll
            for (int ins = 0; ins < 6; ++ins) {
                const int d1 = 2*L1[ins]+1, d2 = 2*L2[ins]+1, d3 = 2*LO[ins]+1;
                const int fbase = OFF_FB + t*FSTR + ((L1[ins] == 4) ? 0 : 9);
                const float* win = (L1[ins] == 4) ? w_in4 : w_in6;
                float* o = (LO[ins] == 4) ? o4 : o6;
                float cw = 0.f;
                #pragma unroll
                for (int uu = 0; uu < 8; ++uu)
                    cw += win[uu] * sm[OFF_CTX + ins*64 + uu*8 + u];
                cw *= (LO[ins] == 4) ? PWV4 : PWV6;
                const float* cgp = &sm[OFF_CTX + 384 + CO[ins]];
                const int shb = (L2[ins] == 0) ? 0 : 1;
                #pragma unroll 1
                for (int i = 0; i < d1; ++i) {
                    const float fi = sm[fbase + i] * cw;
                    const float* cgi = cgp + i*d2*d3;
                    #pragma unroll
                    for (int j = 0; j < d2; ++j) {
                        float fij = fi * shv[shb + j];
                        #pragma unroll
                        for (int k = 0; k < d3; ++k) o[k] += fij * cgi[j*d3 + k];
                    }
                }
            }
            #pragma unroll
            for (int k = 0; k < 9;  ++k) sm[OFF_E + t*DH + u*9 + k]       = o4[k];
            #pragma unroll
            for (int k = 0; k < 13; ++k) sm[OFF_E + t*DH + 72 + u*13 + k] = o6[k];
        }
    }
    __syncthreads();

    // ---- Phase 5: ctx = attn @ vals (64x64 @ 64x176) via WMMA --------------
    for (int tt = wv; tt < 44; tt += 8) {
        const int mt = tt / 11, nt = tt % 11;
        v8f acc = {};
        for (int kk = 0; kk < 16; ++kk) {
            int k0 = kk*4 + lh*2;
            v2f A, Bv;
            A.x  = sm[OFF_SC + (mt*16 + lr)*SSTR + k0];
            A.y  = sm[OFF_SC + (mt*16 + lr)*SSTR + k0 + 1];
            Bv.x = sm[OFF_E + k0*DH       + nt*16 + lr];
            Bv.y = sm[OFF_E + (k0 + 1)*DH + nt*16 + lr];
            acc = __builtin_amdgcn_wmma_f32_16x16x4_f32(false, A, false, Bv,
                                                        (short)0, acc, false, false);
        }
        for (int e = 0; e < 8; ++e)
            sm[OFF_CTX + (mt*16 + e + 8*lh)*DH + nt*16 + lr] = acc[e];
    }
    __syncthreads();

    // ---- Phase 5b: stage tp_out CGs + tp_out weights over dead vals region -
    for (int i = tid; i < CG_OUT_TOTAL; i += 256) sm[OFF_E + i] = cg[CG_VAL_TOTAL + i];
    for (int i = tid; i < 4096;         i += 256) sm[OFF_E + CG_OUT_TOTAL + i] = w_tp_out[i];
    __syncthreads();

    // ---- Phase 6: h_out = TP(h, ctx); project to delta ---------------------
    {
        const float PWO4 = 0.1875f;          // sqrt(9/256)
        const float PWO6 = 0.22534695471f;   // sqrt(13/256)
        const float IS8  = 0.35355339059f;   // 1/sqrt(8)
        for (int task = tid; task < 512; task += 256) {
            const int t = task >> 3, u = task & 7;   // u = output channel
            float o4[9] = {0}, o6[13] = {0};
            const int L1o[8] = {4,4,4,4,6,6,6,6};
            const int L2o[8] = {4,4,6,6,4,4,6,6};
            const int LOo[8] = {4,6,4,6,4,6,4,6};
            const int COo[8] = {0,729,1782,2835,4356,5409,6930,8451};
            #pragma unroll
            for (int ins = 0; ins < 8; ++ins) {
                const int l1 = L1o[ins], l2 = L2o[ins], lo = LOo[ins];
                const int d1 = 2*l1+1, d2 = 2*l2+1, d3 = 2*lo+1;
                const int fbase = OFF_FB + t*FSTR + ((l1 == 4) ? 0 : 9);
                const float* win = (l1 == 4) ? w_in4 : w_in6;
                float* o = (lo == 4) ? o4 : o6;
                const float pw = (lo == 4) ? PWO4 : PWO6;
                const int ctxoff = (l2 == 4) ? 0 : 72;
                // cw[v] = sum_u w_in[u] * W[ins][u][v][uout]
                float cwv[8];
                #pragma unroll
                for (int v = 0; v < 8; ++v) {
                    float s = 0.f;
                    #pragma unroll
                    for (int uu = 0; uu < 8; ++uu)
                        s += win[uu] * sm[OFF_E + CG_OUT_TOTAL + ((ins*8 + uu)*8 + v)*8 + u];
                    cwv[v] = s;
                }
                // d[j] = pw * sum_v cw[v] * ctx[t][v][j]
                float dj[13];
                #pragma unroll
                for (int j = 0; j < d2; ++j) {
                    float s = 0.f;
                    #pragma unroll
                    for (int v = 0; v < 8; ++v)
                        s += cwv[v] * sm[OFF_CTX + t*DH + ctxoff + v*d2 + j];
                    dj[j] = s * pw;
                }
                const float* cgp = &sm[OFF_E + COo[ins]];
                #pragma unroll 1
                for (int i = 0; i < d1; ++i) {
                    const float fi = sm[fbase + i];
                    const float* cgi = cgp + i*d2*d3;
                    #pragma unroll
                    for (int j = 0; j < d2; ++j) {
                        float fd = fi * dj[j];
                        #pragma unroll
                        for (int k = 0; k < d3; ++k) o[k] += fd * cgi[j*d3 + k];
                    }
                }
            }
            // reduce over the 8 output channels (contiguous 8-lane groups share t)
            float wo4 = w_out4[u] * IS8, wo6 = w_out6[u] * IS8;
            #pragma unroll
            for (int i = 0; i < 9; ++i) {
                float v = o4[i] * wo4;
                v += __shfl_xor(v, 1); v += __shfl_xor(v, 2); v += __shfl_xor(v, 4);
                if ((lane & 7) == 0) sm[OFF_DL + t*FSTR + i] = v;
            }
            #pragma unroll
            for (int i = 0; i < 13; ++i) {
                float v = o6[i] * wo6;
                v += __shfl_xor(v, 1); v += __shfl_xor(v, 2); v += __shfl_xor(v, 4);
                if ((lane & 7) == 0) sm[OFF_DL + t*FSTR + 9 + i] = v;
            }
        }
    }
    __syncthreads();

    // ---- Phase 7: scatter delta back to (B, H*W, 22) -----------------------
    for (int i = tid; i < NTOK * 22; i += 256) {
        int t = i / 22, ch = i % 22;
        int n = (by*8 + (t >> 3)) * 128 + (bx*8 + (t & 7));
        out[(size_t)(bimg*16384 + n)*22 + ch] = sm[OFF_DL + t*FSTR + ch];
    }
}

// ---------------------------------------------------------------------------
extern "C" void kernel_launch(void* const* d_in, const int* in_sizes, int n_in,
                              void* d_out, int out_size, void* d_ws, size_t ws_size,
                              hipStream_t stream) {
    const float* feat     = (const float*)d_in[0];
    const float* sh_block = (const float*)d_in[1];
    const float* log_s4   = (const float*)d_in[2];
    const float* log_s6   = (const float*)d_in[3];
    const float* pb_w     = (const float*)d_in[4];
    const float* pb_b     = (const float*)d_in[5];
    const float* w_in4    = (const float*)d_in[6];
    const float* w_in6    = (const float*)d_in[7];
    const float* w_tp_val = (const float*)d_in[8];
    const float* w_tp_out = (const float*)d_in[9];
    const float* w_out4   = (const float*)d_in[10];
    const float* w_out6   = (const float*)d_in[11];
    float* cg = (float*)d_ws;   // 13318 floats of Wigner-3j data
    float* o  = (float*)d_out;

    cg_init_kernel<<<14, 256, 0, stream>>>(cg);

    const size_t smem = (size_t)SMEM_FLOATS * sizeof(float);
    lrblock_attn_kernel<<<1024, 256, smem, stream>>>(
        feat, sh_block, log_s4, log_s6, pb_w, pb_b,
        w_in4, w_in6, w_tp_val, w_tp_out, w_out4, w_out6, cg, o);
}